// Probabilistic_DAG_Generator_17806934409651
// MI455X (gfx1250) — compile-verified
//
#include <hip/hip_runtime.h>
#include <cstdint>
#include <cstddef>

#define TILE 64
#define NTHR 256

#if defined(__HIP_DEVICE_COMPILE__) && __has_builtin(__builtin_amdgcn_global_load_async_to_lds_b128)
#define HAVE_ASYNC_LDS 1
typedef int v4i __attribute__((vector_size(16)));
typedef __attribute__((address_space(1))) v4i v4i_g;   // global int4
typedef __attribute__((address_space(3))) v4i v4i_l;   // LDS int4
#else
#define HAVE_ASYNC_LDS 0
#endif

// Pass 0: nr[j] = 1 - hard_root[j]
__global__ __launch_bounds__(NTHR) void notroot_kernel(
    const float* __restrict__ root_probs,
    const float* __restrict__ g_root,
    float* __restrict__ nr, int n)
{
  int j = blockIdx.x * NTHR + threadIdx.x;
  if (j >= n) return;
  float p  = root_probs[j];
  float g0 = g_root[2 * j + 0];
  float g1 = g_root[2 * j + 1];
  float z0 = p + g0;
  float z1 = (1.0f - p) + g1;
  nr[j] = (z0 > z1) ? 0.0f : 1.0f;
}

// Pass 1: one block per tile pair (bi,bj), bi <= bj.
//   A tile = rows [bi*64), cols [bj*64): all j>i (if bi<bj) -> dagA = uA = s*nr[col]
//   B tile = rows [bj*64), cols [bi*64): all j<i -> dagB = uB * (1 - uA^T)
//   Diagonal tile handles both halves + zero diagonal from one tile in LDS.
__global__ __launch_bounds__(NTHR) void dag_kernel(
    const float* __restrict__ edge,
    const float* __restrict__ ge,
    const float* __restrict__ nr,
    float* __restrict__ out, int n)
{
  const int bi = blockIdx.y;
  const int bj = blockIdx.x;
  if (bi > bj) return;
  const bool diag = (bi == bj);

  __shared__ float uA[TILE][TILE + 1];      // +1 pad -> conflict-free transpose
#if HAVE_ASYNC_LDS
  __shared__ float gB[TILE][2 * TILE];      // async-staged g_edge for mirror tile
#endif

  const int t  = threadIdx.x;
  const int rl = t >> 4;                    // 0..15
  const int c0 = (t & 15) << 2;             // 0,4,...,60 (float4 column)
  const int i0 = bi * TILE;
  const int j0 = bj * TILE;

  const float4 nrA = *(const float4*)(nr + j0 + c0);  // nr over A-tile columns
  const float4 nrB = *(const float4*)(nr + i0 + c0);  // nr over B-tile columns

#if HAVE_ASYNC_LDS
  if (!diag) {
    // Prefetch mirror-tile g_edge (rows j0.., cols i0..) into LDS via the
    // CDNA5 async copy engine while phase 1 computes.
    #pragma unroll
    for (int p = 0; p < 4; ++p) {
      const int r = p * 16 + rl;
      const float* gsrc = ge + 2 * ((size_t)(j0 + r) * n + (i0 + c0));
      __builtin_amdgcn_global_load_async_to_lds_b128(
          (v4i_g*)gsrc,       (v4i_l*)&gB[r][2 * c0],     0, 0);
      __builtin_amdgcn_global_load_async_to_lds_b128(
          (v4i_g*)(gsrc + 4), (v4i_l*)&gB[r][2 * c0 + 4], 0, 0);
    }
  }
#endif

  // ---- Phase 1: A tile ----
  float u_reg[4][4];
  #pragma unroll
  for (int p = 0; p < 4; ++p) {
    const int r = p * 16 + rl;
    const size_t base = (size_t)(i0 + r) * n + (j0 + c0);
    const float4 e4 = *(const float4*)(edge + base);
    const float4 ga = *(const float4*)(ge + 2 * base);
    const float4 gb = *(const float4*)(ge + 2 * base + 4);
    const float e[4]   = {e4.x, e4.y, e4.z, e4.w};
    const float g0v[4] = {ga.x, ga.z, gb.x, gb.z};
    const float g1v[4] = {ga.y, ga.w, gb.y, gb.w};
    const float nrv[4] = {nrA.x, nrA.y, nrA.z, nrA.w};
    #pragma unroll
    for (int k = 0; k < 4; ++k) {
      const float z0 = e[k] + g0v[k];
      const float z1 = (1.0f - e[k]) + g1v[k];
      const float u  = (z0 > z1) ? nrv[k] : 0.0f;   // s * nr[col]
      uA[r][c0 + k] = u;
      u_reg[p][k]   = u;
    }
    if (!diag) {   // upper tile: dag == u, write it out now
      *(float4*)(out + base) =
          make_float4(u_reg[p][0], u_reg[p][1], u_reg[p][2], u_reg[p][3]);
    }
  }

#if HAVE_ASYNC_LDS
  if (!diag) {
    // Each wave drains its own ASYNCcnt before the workgroup barrier so the
    // barrier guarantees all staged data has landed in LDS.
#if __has_builtin(__builtin_amdgcn_s_wait_asynccnt)
    __builtin_amdgcn_s_wait_asynccnt(0);
#else
    asm volatile("s_wait_asynccnt 0" ::: "memory");
#endif
  }
#endif
  __syncthreads();

  // ---- Phase 2 ----
  if (diag) {
    #pragma unroll
    for (int p = 0; p < 4; ++p) {
      const int r = p * 16 + rl;
      float o[4];
      #pragma unroll
      for (int k = 0; k < 4; ++k) {
        const int c = c0 + k;
        float v;
        if (c == r)     v = 0.0f;
        else if (c > r) v = u_reg[p][k];
        else            v = u_reg[p][k] * (1.0f - uA[c][r]);
        o[k] = v;
      }
      *(float4*)(out + (size_t)(i0 + r) * n + (j0 + c0)) =
          make_float4(o[0], o[1], o[2], o[3]);
    }
  } else {
    #pragma unroll
    for (int p = 0; p < 4; ++p) {
      const int r = p * 16 + rl;
      const size_t base = (size_t)(j0 + r) * n + (i0 + c0);
      const float4 e4 = *(const float4*)(edge + base);
      float g0v[4], g1v[4];
#if HAVE_ASYNC_LDS
      #pragma unroll
      for (int k = 0; k < 4; ++k) {
        g0v[k] = gB[r][2 * (c0 + k) + 0];
        g1v[k] = gB[r][2 * (c0 + k) + 1];
      }
#else
      {
        const float4 ga = *(const float4*)(ge + 2 * base);
        const float4 gb = *(const float4*)(ge + 2 * base + 4);
        g0v[0] = ga.x; g1v[0] = ga.y; g0v[1] = ga.z; g1v[1] = ga.w;
        g0v[2] = gb.x; g1v[2] = gb.y; g0v[3] = gb.z; g1v[3] = gb.w;
      }
#endif
      const float e[4]   = {e4.x, e4.y, e4.z, e4.w};
      const float nrv[4] = {nrB.x, nrB.y, nrB.z, nrB.w};
      float o[4];
      #pragma unroll
      for (int k = 0; k < 4; ++k) {
        const float z0 = e[k] + g0v[k];
        const float z1 = (1.0f - e[k]) + g1v[k];
        const float uB = (z0 > z1) ? nrv[k] : 0.0f;
        o[k] = uB * (1.0f - uA[c0 + k][r]);   // conflict-free LDS transpose read
      }
      *(float4*)(out + base) = make_float4(o[0], o[1], o[2], o[3]);
    }
  }
}

extern "C" void kernel_launch(void* const* d_in, const int* in_sizes, int n_in,
                              void* d_out, int out_size, void* d_ws, size_t ws_size,
                              hipStream_t stream) {
  (void)n_in; (void)out_size; (void)ws_size;
  const float* root_probs = (const float*)d_in[0];
  const float* edge_probs = (const float*)d_in[1];
  const float* g_root     = (const float*)d_in[2];
  const float* g_edge     = (const float*)d_in[3];
  float* out = (float*)d_out;
  const int n = in_sizes[0];          // 2048
  float* nr = (float*)d_ws;           // n floats of scratch

  notroot_kernel<<<(n + NTHR - 1) / NTHR, NTHR, 0, stream>>>(
      root_probs, g_root, nr, n);

  const int B = n / TILE;             // 32
  dim3 grid(B, B);                    // blocks with bi > bj exit immediately
  dag_kernel<<<grid, NTHR, 0, stream>>>(edge_probs, g_edge, nr, out, n);
}